// KHopGATLevelPredictor_26104811225561
// MI455X (gfx1250) — compile-verified
//
#include <hip/hip_runtime.h>
#include <hip/hip_bf16.h>

// ---------------------------------------------------------------------------
// KHopGATLevelPredictor on gfx1250 (MI455X)
//   layer GEMMs   : V_WMMA_F32_16X16X4_F32 (fp32 matrix path; GEMM is ~20 GF,
//                   far below the edge-phase memory cost, so keep fp32)
//   edge softmax  : atomicMax on order-preserving uint float encoding,
//                   atomicAdd f32 denominators, wave-per-edge aggregation
// ---------------------------------------------------------------------------

typedef float v2f __attribute__((ext_vector_type(2)));
typedef float v8f __attribute__((ext_vector_type(8)));

#define NNODES 100000
#define NEDGES 600000
#define ETOT   (NEDGES + NNODES)
#define F1     256          // H*C
#define HEADS  4
#define CDIM   64
#define SLOPE  0.2f

__device__ __forceinline__ float lrelu(float e) { return e >= 0.f ? e : SLOPE * e; }

// order-preserving float -> uint encoding for atomicMax
__device__ __forceinline__ unsigned fenc(float f) {
    unsigned u = __float_as_uint(f);
    return (u & 0x80000000u) ? ~u : (u | 0x80000000u);
}
__device__ __forceinline__ float fdec(unsigned k) {
    unsigned u = (k & 0x80000000u) ? (k & 0x7FFFFFFFu) : ~k;
    return __uint_as_float(u);
}

__device__ __forceinline__ void get_edge(const int* __restrict__ ei, int j, int& s, int& d) {
    if (j < NEDGES) { s = ei[j]; d = ei[NEDGES + j]; }
    else            { s = j - NEDGES; d = s; }          // appended self-loops
}

// ---------------------------------------------------------------------------
// GEMM: D[N,256] = A[N,K] @ B[K,256], via V_WMMA_F32_16X16X4_F32.
// grid = (N/16, 4), block = 128 (4 waves); wave w owns cols [by*64+w*16, +16).
// A tile (16 x K) staged in LDS; B streamed (<=256KB total, L2 resident).
// ---------------------------------------------------------------------------
template <int K>
__global__ __launch_bounds__(128) void gemm_wmma_f32(const float* __restrict__ A,
                                                     const float* __restrict__ B,
                                                     float* __restrict__ D) {
    __shared__ float ldsA[16 * K];
    const int tid  = threadIdx.x;
    const int lane = tid & 31;
    const int wave = tid >> 5;
    const int m0   = blockIdx.x * 16;
    const int n0   = blockIdx.y * 64 + wave * 16;

    for (int i = tid; i < 16 * K; i += 128) {
        const int r = i / K, c = i % K;
        ldsA[i] = A[(m0 + r) * K + c];
    }
    __syncthreads();

    v8f acc = {};
    const int mrow  = lane & 15;          // A: lanes 0-15 and 16-31 both hold M=0..15
    const int khalf = (lane >> 4) << 1;   // A/B: upper half-wave holds K+2,K+3
    const int col   = n0 + (lane & 15);

    #pragma unroll 4
    for (int k0 = 0; k0 < K; k0 += 4) {
        const int kk = k0 + khalf;
        v2f a, b;
        a.x = ldsA[mrow * K + kk];
        a.y = ldsA[mrow * K + kk + 1];
        b.x = B[kk * F1 + col];
        b.y = B[(kk + 1) * F1 + col];
        acc = __builtin_amdgcn_wmma_f32_16x16x4_f32(false, a, false, b,
                                                    (short)0, acc, false, false);
    }

    // C/D layout: VGPR r -> (M=r, N=lane) for lanes 0-15, (M=r+8, N=lane-16) else
    const int rbase = (lane < 16) ? 0 : 8;
    #pragma unroll
    for (int r = 0; r < 8; ++r)
        D[(m0 + rbase + r) * F1 + col] = acc[r];
}

// a_src[n,h] = sum_c h[n,h,c]*att_src[h,c] ; a_dst likewise. 1 thread per (n,h).
__global__ void attn_coef(const float* __restrict__ t,
                          const float* __restrict__ att_s,
                          const float* __restrict__ att_d,
                          float* __restrict__ asrc, float* __restrict__ adst) {
    const int i = blockIdx.x * blockDim.x + threadIdx.x;
    if (i >= NNODES * HEADS) return;
    const int n = i >> 2, h = i & 3;
    const float4* tp = (const float4*)(t + n * F1 + h * CDIM);
    const float4* sp = (const float4*)(att_s + h * CDIM);
    const float4* dp = (const float4*)(att_d + h * CDIM);
    float ss = 0.f, sd = 0.f;
    #pragma unroll
    for (int j = 0; j < CDIM / 4; ++j) {
        const float4 v = tp[j], s4 = sp[j], d4 = dp[j];
        ss += v.x * s4.x + v.y * s4.y + v.z * s4.z + v.w * s4.w;
        sd += v.x * d4.x + v.y * d4.y + v.z * d4.z + v.w * d4.w;
    }
    asrc[i] = ss;
    adst[i] = sd;
}

__global__ void zero4(float4* __restrict__ p, int n4) {
    const int i = blockIdx.x * blockDim.x + threadIdx.x;
    if (i < n4) p[i] = make_float4(0.f, 0.f, 0.f, 0.f);
}

// pass 1: segment max (uint-encoded float atomicMax)
__global__ void edge_max(const int* __restrict__ ei,
                         const float* __restrict__ asrc, const float* __restrict__ adst,
                         unsigned* __restrict__ m) {
    const int j = blockIdx.x * blockDim.x + threadIdx.x;
    if (j >= ETOT) return;
    int s, d; get_edge(ei, j, s, d);
    const float4 A  = *(const float4*)(asrc + s * 4);
    const float4 Bv = *(const float4*)(adst + d * 4);
    unsigned* mp = m + d * 4;
    atomicMax(mp + 0, fenc(lrelu(A.x + Bv.x)));
    atomicMax(mp + 1, fenc(lrelu(A.y + Bv.y)));
    atomicMax(mp + 2, fenc(lrelu(A.z + Bv.z)));
    atomicMax(mp + 3, fenc(lrelu(A.w + Bv.w)));
}

// pass 2: denom[dst,h] += exp(e - m[dst,h])
__global__ void edge_sum(const int* __restrict__ ei,
                         const float* __restrict__ asrc, const float* __restrict__ adst,
                         const unsigned* __restrict__ m, float* __restrict__ den) {
    const int j = blockIdx.x * blockDim.x + threadIdx.x;
    if (j >= ETOT) return;
    int s, d; get_edge(ei, j, s, d);
    const float4 A  = *(const float4*)(asrc + s * 4);
    const float4 Bv = *(const float4*)(adst + d * 4);
    const uint4  M  = *(const uint4*)(m + d * 4);
    float* dp = den + d * 4;
    atomicAdd(dp + 0, __expf(lrelu(A.x + Bv.x) - fdec(M.x)));
    atomicAdd(dp + 1, __expf(lrelu(A.y + Bv.y) - fdec(M.y)));
    atomicAdd(dp + 2, __expf(lrelu(A.z + Bv.z) - fdec(M.z)));
    atomicAdd(dp + 3, __expf(lrelu(A.w + Bv.w) - fdec(M.w)));
}

// pass 3: agg[dst, :] += alpha[e,h] * t[src, :]   (wave per edge, lane = stride-32)
__global__ __launch_bounds__(256) void edge_agg(const int* __restrict__ ei,
                                                const float* __restrict__ asrc,
                                                const float* __restrict__ adst,
                                                const unsigned* __restrict__ m,
                                                const float* __restrict__ den,
                                                const float* __restrict__ t,
                                                float* __restrict__ agg) {
    const int gid  = blockIdx.x * blockDim.x + threadIdx.x;
    const int wid  = gid >> 5;
    const int lane = threadIdx.x & 31;
    if (wid >= ETOT) return;
    int s, d; get_edge(ei, wid, s, d);
    const float4 A  = *(const float4*)(asrc + s * 4);
    const float4 Bv = *(const float4*)(adst + d * 4);
    const uint4  M  = *(const uint4*)(m + d * 4);
    const float4 Dn = *(const float4*)(den + d * 4);
    float w[4];
    w[0] = __expf(lrelu(A.x + Bv.x) - fdec(M.x)) / (Dn.x + 1e-16f);
    w[1] = __expf(lrelu(A.y + Bv.y) - fdec(M.y)) / (Dn.y + 1e-16f);
    w[2] = __expf(lrelu(A.z + Bv.z) - fdec(M.z)) / (Dn.z + 1e-16f);
    w[3] = __expf(lrelu(A.w + Bv.w) - fdec(M.w)) / (Dn.w + 1e-16f);
    const float* ts = t + s * F1;
    float* ag = agg + d * F1;
    #pragma unroll
    for (int jj = 0; jj < 8; ++jj) {            // idx>>6 == jj>>1 : uniform head
        const int idx = lane + (jj << 5);
        atomicAdd(ag + idx, w[jj >> 1] * ts[idx]);
    }
}

// h1 = prelu(agg1 + b1), in place
__global__ void post1(float* __restrict__ buf, const float* __restrict__ b1,
                      const float* __restrict__ pa) {
    const int i = blockIdx.x * blockDim.x + threadIdx.x;
    if (i >= NNODES * F1) return;
    const float a = pa[0];
    const float v = buf[i] + b1[i & (F1 - 1)];
    buf[i] = v >= 0.f ? v : a * v;
}

// h2 = prelu(mean_heads(agg2) + b2)
__global__ void post2(const float* __restrict__ agg, const float* __restrict__ b2,
                      const float* __restrict__ pa, float* __restrict__ h2) {
    const int i = blockIdx.x * blockDim.x + threadIdx.x;
    if (i >= NNODES * CDIM) return;
    const int n = i >> 6, c = i & 63;
    const float* p = agg + n * F1 + c;
    const float v = 0.25f * (p[0] + p[64] + p[128] + p[192]) + b2[c];
    const float a = pa[0];
    h2[i] = v >= 0.f ? v : a * v;
}

// out[n] = h2[n,:] @ Wl + bl
__global__ void final_lin(const float* __restrict__ h2, const float* __restrict__ wl,
                          const float* __restrict__ bl, float* __restrict__ out) {
    const int n = blockIdx.x * blockDim.x + threadIdx.x;
    if (n >= NNODES) return;
    const float4* hp = (const float4*)(h2 + n * CDIM);
    const float4* wp = (const float4*)wl;
    float s = 0.f;
    #pragma unroll
    for (int j = 0; j < CDIM / 4; ++j) {
        const float4 h4 = hp[j], w4 = wp[j];
        s += h4.x * w4.x + h4.y * w4.y + h4.z * w4.z + h4.w * w4.w;
    }
    out[n] = s + bl[0];
}

static inline int cdiv(long long a, long long b) { return (int)((a + b - 1) / b); }

extern "C" void kernel_launch(void* const* d_in, const int* in_sizes, int n_in,
                              void* d_out, int out_size, void* d_ws, size_t ws_size,
                              hipStream_t stream) {
    const float* x      = (const float*)d_in[0];
    const int*   ei     = (const int*)  d_in[1];
    const float* W1     = (const float*)d_in[2];
    const float* att_s1 = (const float*)d_in[3];
    const float* att_d1 = (const float*)d_in[4];
    const float* b1     = (const float*)d_in[5];
    const float* W2     = (const float*)d_in[6];
    const float* att_s2 = (const float*)d_in[7];
    const float* att_d2 = (const float*)d_in[8];
    const float* b2     = (const float*)d_in[9];
    const float* pa     = (const float*)d_in[10];
    const float* Wl     = (const float*)d_in[11];
    const float* bl     = (const float*)d_in[12];
    float* out = (float*)d_out;

    // workspace carve-up (all 16B aligned)
    float*    bufA = (float*)d_ws;                 // t1, then t2
    float*    bufB = bufA + (size_t)NNODES * F1;   // agg1 -> h1 -> agg2
    float*    asrc = bufB + (size_t)NNODES * F1;
    float*    adst = asrc + (size_t)NNODES * HEADS;
    unsigned* mbuf = (unsigned*)(adst + (size_t)NNODES * HEADS);
    float*    den  = (float*)(mbuf + (size_t)NNODES * HEADS);
    float*    h2   = den + (size_t)NNODES * HEADS;

    const dim3 gemmGrid(NNODES / 16, F1 / 64);
    const int TB = 256;
    const int gNH   = cdiv((long long)NNODES * HEADS, TB);
    const int gEdge = cdiv(ETOT, TB);
    const int gEw   = cdiv((long long)ETOT * 32, TB);
    const int gNF   = cdiv((long long)NNODES * F1, TB);
    const int gBig4 = cdiv((long long)NNODES * F1 / 4, TB);
    const int gSm4  = cdiv((long long)NNODES * HEADS / 4, TB);

    // ---- layer 1 ----
    gemm_wmma_f32<128><<<gemmGrid, 128, 0, stream>>>(x, W1, bufA);
    attn_coef<<<gNH, TB, 0, stream>>>(bufA, att_s1, att_d1, asrc, adst);
    zero4<<<gBig4, TB, 0, stream>>>((float4*)bufB, NNODES * F1 / 4);
    zero4<<<gSm4, TB, 0, stream>>>((float4*)mbuf, NNODES * HEADS / 4); // 0u = -inf key
    zero4<<<gSm4, TB, 0, stream>>>((float4*)den, NNODES * HEADS / 4);
    edge_max<<<gEdge, TB, 0, stream>>>(ei, asrc, adst, mbuf);
    edge_sum<<<gEdge, TB, 0, stream>>>(ei, asrc, adst, mbuf, den);
    edge_agg<<<gEw, TB, 0, stream>>>(ei, asrc, adst, mbuf, den, bufA, bufB);
    post1<<<gNF, TB, 0, stream>>>(bufB, b1, pa);

    // ---- layer 2 ----
    gemm_wmma_f32<256><<<gemmGrid, 128, 0, stream>>>(bufB, W2, bufA);
    attn_coef<<<gNH, TB, 0, stream>>>(bufA, att_s2, att_d2, asrc, adst);
    zero4<<<gBig4, TB, 0, stream>>>((float4*)bufB, NNODES * F1 / 4);
    zero4<<<gSm4, TB, 0, stream>>>((float4*)mbuf, NNODES * HEADS / 4);
    zero4<<<gSm4, TB, 0, stream>>>((float4*)den, NNODES * HEADS / 4);
    edge_max<<<gEdge, TB, 0, stream>>>(ei, asrc, adst, mbuf);
    edge_sum<<<gEdge, TB, 0, stream>>>(ei, asrc, adst, mbuf, den);
    edge_agg<<<gEw, TB, 0, stream>>>(ei, asrc, adst, mbuf, den, bufA, bufB);
    post2<<<cdiv((long long)NNODES * CDIM, TB), TB, 0, stream>>>(bufB, b2, pa, h2);

    // ---- head ----
    final_lin<<<cdiv(NNODES, TB), TB, 0, stream>>>(h2, Wl, bl, out);
}